// SpanPrunerMulti_85392539779529
// MI455X (gfx1250) — compile-verified
//
#include <hip/hip_runtime.h>

#define BB 8
#define NN 8192
#define DD 256
#define TT 8
#define KK 512
#define NEG_FILLF (-1e20f)

typedef unsigned int v4u __attribute__((ext_vector_type(4)));
typedef int          v4i __attribute__((ext_vector_type(4)));
typedef int          v8i __attribute__((ext_vector_type(8)));

// ------------------------------------------------------------------
// TDM issue helper: 5-arg (ROCm 7.2 / clang-22) vs 6-arg (clang-23+)
// ------------------------------------------------------------------
__device__ __forceinline__ void tdm_gather_load(v4u g0, v8i g1, v4i g2, v4i g3) {
#if defined(__clang_major__) && (__clang_major__ >= 23)
  v8i gz = {0, 0, 0, 0, 0, 0, 0, 0};
  __builtin_amdgcn_tensor_load_to_lds(g0, g1, g2, g3, gz, 0);
#else
  __builtin_amdgcn_tensor_load_to_lds(g0, g1, g2, g3, 0);
#endif
}

// ------------------------------------------------------------------
// Kernel C (placed first so the disasm snippet shows the TDM ops):
// TDM gather-mode row gather of emb0/emb1.
// 256 threads = 8 waves; each wave issues two gather descriptors
// (4 rows x 1KB each tensor) into its private 8KB LDS slice, waits
// on TENSORcnt, then streams LDS -> d_out with b128 ops.
// Grid: 128 blocks (16 per batch, 32 rows per block).
// ------------------------------------------------------------------
__global__ __launch_bounds__(256) void gather_kernel(
    const float* __restrict__ emb0, const float* __restrict__ emb1,
    const int* __restrict__ widx,
    float* __restrict__ out0, float* __restrict__ out1)
{
  __shared__ float gbuf[16384];                        // 64KB, 8KB per wave
  const int tid  = threadIdx.x;
  const int w    = tid >> 5;
  const int lane = tid & 31;
  const int b    = blockIdx.x >> 4;
  const int k0   = (blockIdx.x & 15) * 32;
  const int kw   = k0 + w * 4;                         // first of 4 rows for this wave

  int myidx = 0;
  if (lane < 4) myidx = widx[b * KK + kw + lane];
  const int i0 = __builtin_amdgcn_readlane(myidx, 0);
  const int i1 = __builtin_amdgcn_readlane(myidx, 1);
  const int i2 = __builtin_amdgcn_readlane(myidx, 2);
  const int i3 = __builtin_amdgcn_readlane(myidx, 3);

  // D# group2/3: 16-bit gather row indices (4 valid)
  v4i g2 = { (int)((i0 & 0xFFFF) | (i1 << 16)),
             (int)((i2 & 0xFFFF) | (i3 << 16)), 0, 0 };
  v4i g3 = { 0, 0, 0, 0 };

  // D# group1: data_size=4B, tensor 256x8192, tile 256x4, stride 256
  v8i g1 = { 0x00020000,            // wg_mask=0 | data_size=2 (4B)
             (int)(DD << 16),       // tensor_dim0[15:0]=256
             (int)(NN << 16),       // tensor_dim0 hi=0 | tensor_dim1[15:0]=8192
             (int)(DD << 16),       // tensor_dim1 hi=0 | tile_dim0=256
             4,                     // tile_dim1 = 4 valid indices
             DD,                    // tensor_dim0_stride = 256
             0, 0 };

  {
    unsigned long long ga = (unsigned long long)(uintptr_t)(emb0 + (size_t)b * NN * DD);
    v4u g0 = { 0x80000001u,                                   // count=1, gather_mode=1, 16b idx
               (unsigned)(w * 8192),                          // lds_addr (bytes)
               (unsigned)(ga & 0xFFFFFFFFu),
               (unsigned)((ga >> 32) & 0x01FFFFFFu) | 0x80000000u };  // type=2
    tdm_gather_load(g0, g1, g2, g3);
  }
  {
    unsigned long long ga = (unsigned long long)(uintptr_t)(emb1 + (size_t)b * NN * DD);
    v4u g0 = { 0x80000001u,
               (unsigned)(w * 8192 + 4096),
               (unsigned)(ga & 0xFFFFFFFFu),
               (unsigned)((ga >> 32) & 0x01FFFFFFu) | 0x80000000u };
    tdm_gather_load(g0, g1, g2, g3);
  }

  __builtin_amdgcn_s_wait_tensorcnt(0);

  // Drain this wave's LDS slice: 4 rows x 256 f32 per tensor.
  const float4* lds4 = (const float4*)gbuf;
  float4* o0 = (float4*)(out0 + (size_t)(b * KK + kw) * DD);
  float4* o1 = (float4*)(out1 + (size_t)(b * KK + kw) * DD);
  for (int t = lane; t < 256; t += 32) {
    o0[t] = lds4[w * 512 + t];
    o1[t] = lds4[w * 512 + 256 + t];
  }
}

// ------------------------------------------------------------------
// Kernel A: masked probabilities  (p = max tail-softmax of 8 scores)
// ------------------------------------------------------------------
__device__ __forceinline__ float tail_sm_max(const float v[TT]) {
  float m = v[0];
#pragma unroll
  for (int t = 1; t < TT; ++t) m = fmaxf(m, v[t]);
  float s = 0.f;
#pragma unroll
  for (int t = 0; t < TT; ++t) s += __expf(v[t] - m);
  float m1 = v[1];
#pragma unroll
  for (int t = 2; t < TT; ++t) m1 = fmaxf(m1, v[t]);
  return __expf(m1 - m) / s;
}

__global__ __launch_bounds__(256) void span_scores_kernel(
    const float* __restrict__ s0, const float* __restrict__ s1,
    const int* __restrict__ span_mask, float* __restrict__ msk)
{
  int i = blockIdx.x * 256 + threadIdx.x;
  if (i >= BB * NN) return;
  float v[TT];
  {
    const float4* p = (const float4*)(s0 + (size_t)i * TT);
    float4 a = p[0], b = p[1];
    v[0] = a.x; v[1] = a.y; v[2] = a.z; v[3] = a.w;
    v[4] = b.x; v[5] = b.y; v[6] = b.z; v[7] = b.w;
  }
  float p0 = tail_sm_max(v);
  {
    const float4* p = (const float4*)(s1 + (size_t)i * TT);
    float4 a = p[0], b = p[1];
    v[0] = a.x; v[1] = a.y; v[2] = a.z; v[3] = a.w;
    v[4] = b.x; v[5] = b.y; v[6] = b.z; v[7] = b.w;
  }
  float p1 = tail_sm_max(v);
  float prob = fmaxf(p0, p1);
  msk[i] = (span_mask[i] != 0) ? prob : NEG_FILLF;
}

// ------------------------------------------------------------------
// Kernel B: per-batch top-512 + fill + ascending index sort + outputs
// One 1024-thread workgroup per batch; 64KB LDS bitonic sort of
// packed keys (monotone(score)<<32 | ~index) = descending score,
// ties broken by smaller index, exactly like jax.lax.top_k.
// ------------------------------------------------------------------
__global__ __launch_bounds__(1024) void topk_kernel(
    const float* __restrict__ msk, const int* __restrict__ span_mask,
    const int* __restrict__ seq_len,
    float* __restrict__ o_idx, float* __restrict__ o_sc,
    float* __restrict__ o_mk, int* __restrict__ widx)
{
  __shared__ unsigned long long keys[NN];
  __shared__ int sfill;
  const int b = blockIdx.x;
  const int tid = threadIdx.x;
  const float* ms = msk + (size_t)b * NN;

  for (int i = tid; i < NN; i += 1024) {
    unsigned u = __float_as_uint(ms[i]);
    u = (u & 0x80000000u) ? ~u : (u | 0x80000000u);   // order-preserving map
    keys[i] = ((unsigned long long)u << 32) | (unsigned)(~i);
  }
  if (tid == 0) sfill = -1;
  __syncthreads();

  // bitonic sort, descending
  for (int k = 2; k <= NN; k <<= 1) {
    for (int j = k >> 1; j > 0; j >>= 1) {
      for (int i = tid; i < NN; i += 1024) {
        int p = i ^ j;
        if (p > i) {
          bool desc = ((i & k) == 0);
          unsigned long long a = keys[i], c = keys[p];
          if ((a < c) == desc) { keys[i] = c; keys[p] = a; }
        }
      }
      __syncthreads();
    }
  }

  int nk = seq_len[b];
  if (nk < 1) nk = 1;
  if (nk > NN) nk = NN;
  if (nk > KK) nk = KK;

  int idx_t = 0;
  if (tid < KK) {
    idx_t = (int)(~(unsigned)keys[tid]);              // recover index
    atomicMax(&sfill, idx_t);                          // fill = max of top-512
  }
  __syncthreads();
  const int fill = sfill;

  int* ik = (int*)keys;                                // reuse LDS
  if (tid < KK) ik[tid] = (tid < nk) ? idx_t : fill;
  __syncthreads();

  // ascending bitonic sort of 512 indices
  for (int k = 2; k <= KK; k <<= 1) {
    for (int j = k >> 1; j > 0; j >>= 1) {
      if (tid < KK) {
        int p = tid ^ j;
        if (p > tid) {
          bool asc = ((tid & k) == 0);
          int a = ik[tid], c = ik[p];
          if ((a > c) == asc) { ik[tid] = c; ik[p] = a; }
        }
      }
      __syncthreads();
    }
  }

  if (tid < KK) {
    int s = ik[tid];
    o_idx[b * KK + tid] = (float)s;
    o_sc [b * KK + tid] = ms[s];
    bool sm = span_mask[(size_t)b * NN + s] != 0;
    o_mk [b * KK + tid] = ((tid < nk) && sm) ? 1.0f : 0.0f;
    widx [b * KK + tid] = s;
  }
}

// ------------------------------------------------------------------
extern "C" void kernel_launch(void* const* d_in, const int* in_sizes, int n_in,
                              void* d_out, int out_size, void* d_ws, size_t ws_size,
                              hipStream_t stream) {
  const float* emb0     = (const float*)d_in[0];
  const float* emb1     = (const float*)d_in[1];
  const float* scores0  = (const float*)d_in[2];
  const float* scores1  = (const float*)d_in[3];
  const int*   span_msk = (const int*)d_in[4];
  const int*   seq_len  = (const int*)d_in[5];

  float* out   = (float*)d_out;
  float* o_idx = out;                            // B*K
  float* o_e0  = o_idx + (size_t)BB * KK;        // B*K*D
  float* o_e1  = o_e0  + (size_t)BB * KK * DD;   // B*K*D
  float* o_sc  = o_e1  + (size_t)BB * KK * DD;   // B*K
  float* o_mk  = o_sc  + (size_t)BB * KK;        // B*K

  float* msk  = (float*)d_ws;                    // B*N masked probabilities
  int*   widx = (int*)(msk + (size_t)BB * NN);   // B*K selected indices

  span_scores_kernel<<<(BB * NN) / 256, 256, 0, stream>>>(scores0, scores1, span_msk, msk);
  topk_kernel<<<BB, 1024, 0, stream>>>(msk, span_msk, seq_len, o_idx, o_sc, o_mk, widx);
  gather_kernel<<<(BB * KK) / 32, 256, 0, stream>>>(emb0, emb1, widx, o_e0, o_e1);
}